// GradientLoss_91061896609894
// MI455X (gfx1250) — compile-verified
//
#include <hip/hip_runtime.h>
#include <math.h>
#include <stdint.h>

// Volume geometry (fixed by the reference: [B=2, C=1, D=160, H=192, W=160])
#define Wd 160
#define Hd 192
#define Dd 160
#define Bn 2
// Tiling: 32x8 threads in (x,y), rolling 3-plane LDS buffer along z, z chunks of 20.
#define TX 32
#define TY 8
#define ZC 20
#define ZCHUNKS (Dd / ZC)       // 8
#define LXS (TX + 2)            // 34
#define LYS (TY + 2)            // 10
#define PLANE (LXS * LYS)       // 340 floats per plane per tensor
#define NTHREADS (TX * TY)      // 256 = 8 wave32
#define TAIL (PLANE - NTHREADS) // 84

typedef float v2f __attribute__((ext_vector_type(2)));
typedef float v8f __attribute__((ext_vector_type(8)));
typedef unsigned int u32x4 __attribute__((ext_vector_type(4)));
typedef int i32x8 __attribute__((ext_vector_type(8)));
typedef int i32x4 __attribute__((ext_vector_type(4)));

#if defined(__gfx1250__) && __has_builtin(__builtin_amdgcn_tensor_load_to_lds) && \
    __has_builtin(__builtin_amdgcn_s_wait_tensorcnt)
#define USE_TDM 1
#else
#define USE_TDM 0
#endif

__device__ __forceinline__ int slot3(int z) { return (z + 3) % 3; }  // valid for z >= -3

#if USE_TDM
// Issue one TDM descriptor: 34x10 fp32 2D tile starting at (z, y0-1, x0-1).
// Caller guarantees the tile is fully in-bounds (interior tile, z in [0,D)).
// D# layout per CDNA5 ISA ch.8 (group0: count/lds_addr/global_addr/type=2,
// group1: data_size, tensor dims from tile origin, tile dims, dim0 stride).
// This toolchain exposes the 6-arg builtin:
//   (u32x4 g0, i32x8 g1, i32x4 g2, i32x4 g3, i32x8 aux, i32 cpol)
__device__ __forceinline__ void tdm_load_plane(const float* __restrict__ src,
                                               uint32_t lds_off,
                                               int b, int z, int x0, int y0) {
    uint64_t ga = (uint64_t)(uintptr_t)(src +
        (((size_t)(b * Dd + z) * Hd + (size_t)(y0 - 1)) * Wd + (size_t)(x0 - 1)));
    uint32_t td0 = (uint32_t)(Wd - (x0 - 1));   // valid elems along dim0 from tile origin
    uint32_t td1 = (uint32_t)(Hd - (y0 - 1));   // valid lines along dim1 from tile origin
    u32x4 g0;
    g0[0] = 1u;                                  // count=1 (valid), user mode, no gather
    g0[1] = lds_off;                             // LDS byte address of tile
    g0[2] = (uint32_t)ga;                        // global_addr[31:0]
    g0[3] = (uint32_t)((ga >> 32) & 0x01ffffffu) // global_addr[56:32]
            | (2u << 30);                        // type = 2 ("image")
    i32x8 g1;
    g1[0] = (int)(2u << 16);                     // workgroup_mask=0, data_size=2 (4 bytes)
    g1[1] = (int)((td0 & 0xffffu) << 16);        // atomic_barrier=0, tensor_dim0[15:0]
    g1[2] = (int)(((td0 >> 16) & 0xffffu) | ((td1 & 0xffffu) << 16));   // td0 hi, td1 lo
    g1[3] = (int)(((td1 >> 16) & 0xffffu) | ((uint32_t)LXS << 16));     // td1 hi, tile_dim0=34
    g1[4] = (int)LYS;                            // tile_dim1=10, tile_dim2=0
    g1[5] = (int)Wd;                             // tensor_dim0_stride = W
    g1[6] = 0;                                   // stride0 hi, tensor_dim1_stride lo (unused)
    g1[7] = 0;
    i32x4 gz4 = {0, 0, 0, 0};                    // groups 2/3 unused (2D tile)
    i32x8 gz8 = {0, 0, 0, 0, 0, 0, 0, 0};        // aux group (zero-filled)
    __builtin_amdgcn_tensor_load_to_lds(g0, g1, gz4, gz4, gz8, 0);
}
#endif

// Branchless halo load of one element (clamped address + select-zero padding).
__device__ __forceinline__ void load_elem(const float* __restrict__ pred,
                                          const float* __restrict__ targ,
                                          float* __restrict__ smP,
                                          float* __restrict__ smT,
                                          int b, int z, int x0, int y0, int i) {
    int ly = i / LXS;
    int lx = i - ly * LXS;
    int y = y0 - 1 + ly;
    int x = x0 - 1 + lx;
    bool in = (z >= 0) && (z < Dd) && ((unsigned)y < (unsigned)Hd) &&
              ((unsigned)x < (unsigned)Wd);
    int zc = min(max(z, 0), Dd - 1);
    int yc = min(max(y, 0), Hd - 1);
    int xc = min(max(x, 0), Wd - 1);
    int idx = ((b * Dd + zc) * Hd + yc) * Wd + xc;
    float vp = pred[idx];                        // unconditional (clamped) loads
    float vt = targ[idx];
    smP[i] = in ? vp : 0.0f;                     // zero padding via v_cndmask
    smT[i] = in ? vt : 0.0f;
}

// Separable 3D Sobel gradient magnitude from 3 LDS planes (z-1, z, z+1).
// kx = smoothZ (x) smoothY (x) derivX, etc.; smooth=[1,2,1], deriv=[-1,0,1].
__device__ __forceinline__ float gmag3(const float* __restrict__ p0,
                                       const float* __restrict__ p1,
                                       const float* __restrict__ p2,
                                       int off) {
    const float* pls[3] = {p0, p1, p2};
    float A[3], Bv[3], Cv[3];
    #pragma unroll
    for (int iz = 0; iz < 3; ++iz) {
        float s[3], dd[3];
        #pragma unroll
        for (int iy = 0; iy < 3; ++iy) {
            const float* r = pls[iz] + off + (iy - 1) * LXS;
            float vm = r[-1], v0 = r[0], vp = r[1];
            s[iy]  = vm + 2.0f * v0 + vp;   // x-smooth
            dd[iy] = vp - vm;               // x-derivative
        }
        A[iz]  = dd[0] + 2.0f * dd[1] + dd[2];  // y-smooth of x-deriv
        Bv[iz] = s[2] - s[0];                   // y-deriv  of x-smooth
        Cv[iz] = s[0] + 2.0f * s[1] + s[2];     // y-smooth of x-smooth
    }
    float gx = A[0]  + 2.0f * A[1]  + A[2];
    float gy = Bv[0] + 2.0f * Bv[1] + Bv[2];
    float gz = Cv[2] - Cv[0];
    return sqrtf(gx * gx + gy * gy + gz * gz + 1e-8f);
}

__global__ __launch_bounds__(NTHREADS)
void sobel_loss_kernel(const float* __restrict__ pred,
                       const float* __restrict__ targ,
                       float* __restrict__ partials) {
    __shared__ float smP[3][PLANE];
    __shared__ float smT[3][PLANE];
    __shared__ float wsum[NTHREADS / 32];

    const int tx = threadIdx.x, ty = threadIdx.y;
    const int tid = ty * TX + tx;
    const int x0 = blockIdx.x * TX;
    const int y0 = blockIdx.y * TY;
    const int b  = blockIdx.z / ZCHUNKS;
    const int zc = blockIdx.z % ZCHUNKS;
    const int z_start = zc * ZC;
    const int off = (ty + 1) * LXS + (tx + 1);
    // Tile fully interior in x/y (halo never leaves the volume)?
    const bool xy_int = (x0 > 0) && (x0 + TX < Wd) && (y0 > 0) && (y0 + TY < Hd);

    // ---- plane loader (dispatch: zero-fill / TDM DMA / branchless manual) ----
    auto load_plane = [&](int z) {
        const int sl = slot3(z);
        if (z < 0 || z >= Dd) {
            // zero-padding plane outside the volume
            smP[sl][tid] = 0.0f;
            smT[sl][tid] = 0.0f;
            if (tid < TAIL) {
                smP[sl][tid + NTHREADS] = 0.0f;
                smT[sl][tid + NTHREADS] = 0.0f;
            }
            return;
        }
#if USE_TDM
        if (xy_int) {
            // One wave drives the Tensor Data Mover; DMA deposits both 34x10
            // tiles straight into LDS; TENSORcnt guards completion before the
            // workgroup barrier makes the data visible.
            if (tid < 32) {
                tdm_load_plane(pred, (uint32_t)(uintptr_t)(void*)&smP[sl][0],
                               b, z, x0, y0);
                tdm_load_plane(targ, (uint32_t)(uintptr_t)(void*)&smT[sl][0],
                               b, z, x0, y0);
                __builtin_amdgcn_s_wait_tensorcnt((short)0);
            }
            return;
        }
#endif
        load_elem(pred, targ, smP[sl], smT[sl], b, z, x0, y0, tid);
        if (tid < TAIL)
            load_elem(pred, targ, smP[sl], smT[sl], b, z, x0, y0, tid + NTHREADS);
    };

    // Preload planes z_start-1 and z_start.
    load_plane(z_start - 1);
    load_plane(z_start);

    float acc = 0.0f;
    for (int z = z_start; z < z_start + ZC; ++z) {
        __syncthreads();  // prior compute done (slot z-2 free) + preloads visible
        // Prefetch plane z+2 into cache while we work on plane z.
        int zp = z + 2;
        if (zp < Dd) {
            int pidx = ((b * Dd + zp) * Hd + (y0 + ty)) * Wd + (x0 + tx);
            __builtin_prefetch(pred + pidx, 0, 0);
            __builtin_prefetch(targ + pidx, 0, 0);
        }
        load_plane(z + 1);
        __syncthreads();  // plane z+1 visible to all waves
        float pg = gmag3(smP[slot3(z - 1)], smP[slot3(z)], smP[slot3(z + 1)], off);
        float tg = gmag3(smT[slot3(z - 1)], smT[slot3(z)], smT[slot3(z + 1)], off);
        acc += fabsf(pg - tg);
    }

    // ---- Wave-level reduction via V_WMMA_F32_16X16X4_F32 ----
    // A (16x4 f32) = per-lane accumulators (VGPR1 = 0); B (4x16) = ones.
    // D[m][n] = sum_k A[m][k] -> row sums; lanes 0..15 hold rows 0..7 in the
    // 8 C VGPRs, lanes 16..31 hold rows 8..15; one xor-16 shuffle finishes.
    v2f a; a[0] = acc;  a[1] = 0.0f;
    v2f bo; bo[0] = 1.0f; bo[1] = 1.0f;
    v8f c = {};
    c = __builtin_amdgcn_wmma_f32_16x16x4_f32(
            /*neg_a=*/false, a, /*neg_b=*/false, bo,
            /*c_mod=*/(short)0, c, /*reuse_a=*/false, /*reuse_b=*/false);
    float s = c[0] + c[1] + c[2] + c[3] + c[4] + c[5] + c[6] + c[7];
    s += __shfl_xor(s, 16, 32);

    const int lane = tid & 31, wave = tid >> 5;
    if (lane == 0) wsum[wave] = s;
    __syncthreads();
    if (tid == 0) {
        float t = 0.0f;
        #pragma unroll
        for (int i = 0; i < NTHREADS / 32; ++i) t += wsum[i];
        int blockId = blockIdx.x + gridDim.x * (blockIdx.y + gridDim.y * blockIdx.z);
        partials[blockId] = t;
    }
}

__global__ __launch_bounds__(256)
void final_reduce_kernel(const float* __restrict__ partials, int n, float invN,
                         float* __restrict__ out) {
    __shared__ float red[256];
    float s = 0.0f;
    for (int i = threadIdx.x; i < n; i += 256) s += partials[i];
    red[threadIdx.x] = s;
    __syncthreads();
    for (int k = 128; k > 0; k >>= 1) {
        if ((int)threadIdx.x < k) red[threadIdx.x] += red[threadIdx.x + k];
        __syncthreads();
    }
    if (threadIdx.x == 0) out[0] = red[0] * invN;
}

extern "C" void kernel_launch(void* const* d_in, const int* in_sizes, int n_in,
                              void* d_out, int out_size, void* d_ws, size_t ws_size,
                              hipStream_t stream) {
    const float* pred = (const float*)d_in[0];
    const float* targ = (const float*)d_in[1];
    float* out = (float*)d_out;
    float* partials = (float*)d_ws;

    dim3 block(TX, TY, 1);
    dim3 grid(Wd / TX, Hd / TY, Bn * ZCHUNKS);   // 5 x 24 x 16 = 1920 blocks
    sobel_loss_kernel<<<grid, block, 0, stream>>>(pred, targ, partials);

    const int nb = (Wd / TX) * (Hd / TY) * (Bn * ZCHUNKS);
    const float invN = 1.0f / (float)((long)Bn * Dd * Hd * Wd);
    final_reduce_kernel<<<1, 256, 0, stream>>>(partials, nb, invN, out);
}